// Grid4DSpatiotemporalEncoder_27943057227891
// MI455X (gfx1250) — compile-verified
//
#include <hip/hip_runtime.h>
#include <math.h>

#define NPTS       262144
#define SP_LEVELS  24
#define SP_F       48           // 24 levels * 2 feats
#define T_LEVELS   19
#define T_F        114          // 3 planes * 19 levels * 2 feats
#define T_PLANE_F  38

typedef float v2f __attribute__((ext_vector_type(2)));

struct SMeta { int res[SP_LEVELS]; int off[SP_LEVELS]; int n_dense; };
struct TMeta { int r0[T_LEVELS]; int r1[T_LEVELS]; int r2[T_LEVELS]; int off[T_LEVELS]; };

__device__ __forceinline__ float clamp01(float v) {
    return fminf(fmaxf(v, 0.0f), 1.0f);
}

// 8-corner trilinear accumulate; corner i: bit2->dim0, bit1->dim1, bit0->dim2
__device__ __forceinline__ float2 interp8(const v2f f[8], float wa, float wb, float wc) {
    float w0a = 1.0f - wa, w0b = 1.0f - wb, w0c = 1.0f - wc;
    float wgt[8] = { w0a*w0b*w0c, w0a*w0b*wc, w0a*wb*w0c, w0a*wb*wc,
                     wa*w0b*w0c,  wa*w0b*wc,  wa*wb*w0c,  wa*wb*wc };
    float o0 = 0.0f, o1 = 0.0f;
    #pragma unroll
    for (int k = 0; k < 8; ++k) {
        o0 = fmaf(wgt[k], f[k].x, o0);
        o1 = fmaf(wgt[k], f[k].y, o1);
    }
    return make_float2(o0, o1);
}

// grid.y = 4: y==0 -> spatial (24-level 3D hash grid), y==1..3 -> temporal planes.
// Spatial waves are HBM-latency-bound, temporal waves are L2-hit-bound; mixing
// them in one dispatch lets each SIMD overlap the two stall profiles.
__global__ __launch_bounds__(256) void grid4d_kernel(
    const float4* __restrict__ xyzt,
    const float* __restrict__ xyz_t,
    const float* __restrict__ xyt_t, const float* __restrict__ yzt_t,
    const float* __restrict__ xzt_t,
    float* __restrict__ out, SMeta sm, TMeta tm)
{
    int p = blockIdx.x * blockDim.x + threadIdx.x;
    if (p >= NPTS) return;
    float4 c = xyzt[p];
    int part = blockIdx.y;

    if (part == 0) {
        // ----------------- spatial -----------------
        float ux = clamp01((c.x + 1.0f) * 0.5f);
        float uy = clamp01((c.y + 1.0f) * 0.5f);
        float uz = clamp01((c.z + 1.0f) * 0.5f);
        float2* orow = (float2*)(out + (size_t)p * SP_F);

        // dense low-res levels (cells <= 2^22)
        for (int l = 0; l < sm.n_dense; ++l) {
            int   res = sm.res[l];
            float sc  = (float)(res - 1);
            float hi  = (float)(res - 2 > 0 ? res - 2 : 0);
            float px = ux * sc, py = uy * sc, pz = uz * sc;
            float fx = fminf(fmaxf(floorf(px), 0.0f), hi);
            float fy = fminf(fmaxf(floorf(py), 0.0f), hi);
            float fz = fminf(fmaxf(floorf(pz), 0.0f), hi);
            int ix = (int)fx, iy = (int)fy, iz = (int)fz;
            int s1 = res, s0 = res * res;
            int b = ix * s0 + iy * s1 + iz;
            const v2f* tb2 = (const v2f*)(xyz_t + (size_t)sm.off[l] * 2);
            int idx[8] = { b,           b + 1,
                           b + s1,      b + s1 + 1,
                           b + s0,      b + s0 + 1,
                           b + s0 + s1, b + s0 + s1 + 1 };
            v2f f[8];
            #pragma unroll
            for (int k = 0; k < 8; ++k) f[k] = tb2[idx[k]];
            orow[l] = interp8(f, px - fx, py - fy, pz - fz);
        }

        // hashed high-res levels: unroll x2 -> 16 outstanding HBM gathers/wave
        #pragma unroll 2
        for (int l = sm.n_dense; l < SP_LEVELS; ++l) {
            int   res = sm.res[l];
            float sc  = (float)(res - 1);
            float hi  = (float)(res - 2);
            float px = ux * sc, py = uy * sc, pz = uz * sc;
            float fx = fminf(fmaxf(floorf(px), 0.0f), hi);
            float fy = fminf(fmaxf(floorf(py), 0.0f), hi);
            float fz = fminf(fmaxf(floorf(pz), 0.0f), hi);
            unsigned hx0 = (unsigned)(int)fx;
            unsigned hy0 = (unsigned)(int)fy * 2654435761u;
            unsigned hz0 = (unsigned)(int)fz * 805459861u;
            unsigned hx1 = hx0 + 1u;
            unsigned hy1 = hy0 + 2654435761u;
            unsigned hz1 = hz0 + 805459861u;
            const unsigned mask = (1u << 22) - 1u;
            unsigned idx[8] = { (hx0^hy0^hz0) & mask, (hx0^hy0^hz1) & mask,
                                (hx0^hy1^hz0) & mask, (hx0^hy1^hz1) & mask,
                                (hx1^hy0^hz0) & mask, (hx1^hy0^hz1) & mask,
                                (hx1^hy1^hz0) & mask, (hx1^hy1^hz1) & mask };
            const v2f* tb2 = (const v2f*)(xyz_t + (size_t)sm.off[l] * 2);
            v2f f[8];
            // 690MB streaming-random table: non-temporal so it doesn't evict
            // the hot temporal tables / write-combining lines from L2
            #pragma unroll
            for (int k = 0; k < 8; ++k)
                f[k] = __builtin_nontemporal_load(tb2 + idx[k]);
            orow[l] = interp8(f, px - fx, py - fy, pz - fz);
        }
    } else {
        // ----------------- temporal plane -----------------
        int pl = part - 1;                 // 0:xyt 1:yzt 2:xzt
        float t  = (c.w * 2.0f - 1.0f) * 0.9f;
        float ux = clamp01((c.x + 1.0f) * 0.5f);
        float uy = clamp01((c.y + 1.0f) * 0.5f);
        float uz = clamp01((c.z + 1.0f) * 0.5f);
        float ut = clamp01((t   + 1.0f) * 0.5f);
        float ua = (pl == 1) ? uy : ux;    // xyt:x  yzt:y  xzt:x
        float ub = (pl == 0) ? uy : uz;    // xyt:y  yzt:z  xzt:z
        const float* table = (pl == 0) ? xyt_t : (pl == 1) ? yzt_t : xzt_t;
        float2* orow = (float2*)(out + (size_t)NPTS * SP_F
                                     + (size_t)p * T_F + pl * T_PLANE_F);

        for (int l = 0; l < T_LEVELS; ++l) {
            int r0 = tm.r0[l], r1 = tm.r1[l], r2 = tm.r2[l];
            float pa = ua * (float)(r0 - 1);
            float pb = ub * (float)(r1 - 1);
            float pt = ut * (float)(r2 - 1);
            float fa = fminf(fmaxf(floorf(pa), 0.0f), (float)(r0 - 2 > 0 ? r0 - 2 : 0));
            float fb = fminf(fmaxf(floorf(pb), 0.0f), (float)(r1 - 2 > 0 ? r1 - 2 : 0));
            float ft = fminf(fmaxf(floorf(pt), 0.0f), (float)(r2 - 2 > 0 ? r2 - 2 : 0));
            int ia = (int)fa, ib = (int)fb, it = (int)ft;
            int s1 = r2, s0 = r1 * r2;
            int b  = ia * s0 + ib * s1 + it;
            int idx[8] = { b,           b + 1,
                           b + s1,      b + s1 + 1,
                           b + s0,      b + s0 + 1,
                           b + s0 + s1, b + s0 + s1 + 1 };
            const v2f* tb2 = (const v2f*)(table + (size_t)tm.off[l] * 2);
            v2f f[8];
            #pragma unroll
            for (int k = 0; k < 8; ++k) f[k] = tb2[idx[k]];
            orow[l] = interp8(f, pa - fa, pb - fb, pt - ft);
        }
    }
}

extern "C" void kernel_launch(void* const* d_in, const int* in_sizes, int n_in,
                              void* d_out, int out_size, void* d_ws, size_t ws_size,
                              hipStream_t stream) {
    const float4* xyzt  = (const float4*)d_in[0];
    const float*  xyz_t = (const float*)d_in[1];
    const float*  xyt_t = (const float*)d_in[2];
    const float*  yzt_t = (const float*)d_in[3];
    const float*  xzt_t = (const float*)d_in[4];
    float* out = (float*)d_out;

    // Recreate build_meta([16]*3, [2^27]*3, 24, 22) in float64 (ref: ceil(base*scale^l - 1e-6))
    SMeta sm;
    {
        double base = 16.0, desired = 134217728.0;
        double scale = exp((log(desired) - log(base)) / 23.0);
        long long off = 0, hash = 1ll << 22;
        int nd = 0;
        for (int l = 0; l < SP_LEVELS; ++l) {
            long long r = (long long)ceil(base * pow(scale, (double)l) - 1e-6);
            long long cells = r * r * r;
            int uh = cells > hash;
            sm.res[l] = (int)r;
            sm.off[l] = (int)off;
            if (!uh) nd = l + 1;          // use_hash is monotone in level
            off += uh ? hash : cells;
        }
        sm.n_dense = nd;
    }
    // Recreate build_meta([8]*3, [32,32,16], 19, 18): all levels dense
    TMeta tm;
    {
        double base = 8.0;
        double sA = exp((log(32.0) - log(base)) / 18.0);   // dims 0,1
        double sT = exp((log(16.0) - log(base)) / 18.0);   // dim 2
        long long off = 0, hash = 1ll << 18;
        for (int l = 0; l < T_LEVELS; ++l) {
            long long r0 = (long long)ceil(base * pow(sA, (double)l) - 1e-6);
            long long r2 = (long long)ceil(base * pow(sT, (double)l) - 1e-6);
            long long cells = r0 * r0 * r2;
            tm.r0[l] = (int)r0; tm.r1[l] = (int)r0; tm.r2[l] = (int)r2;
            tm.off[l] = (int)off;
            off += (cells > hash) ? hash : cells;
        }
    }

    dim3 blk(256);
    dim3 grid((NPTS + 255) / 256, 4);   // y=0 spatial, y=1..3 temporal planes
    grid4d_kernel<<<grid, blk, 0, stream>>>(xyzt, xyz_t, xyt_t, yzt_t, xzt_t,
                                            out, sm, tm);
}